// StyleNet_90460601188411
// MI455X (gfx1250) — compile-verified
//
#include <hip/hip_runtime.h>

typedef float v2f __attribute__((ext_vector_type(2)));
typedef float v8f __attribute__((ext_vector_type(8)));

#define B_SAMPLES   16384
#define AE_DIM      2048
#define STYLE_DIM   128
#define NUM_DOMAINS 8
#define BM          64   // rows per workgroup (4x fewer W re-reads from L2)
#define BK          64
#define A_STRIDE    68   // 16B-aligned float4 stores, conflict-free frag reads
#define B_STRIDE    68

// ---- ws layout (ints): [0..7] counts | [8..15] cursors | [16..16+B) bucketed indices ----

__global__ void zero_counters_kernel(int* ws) {
    if (threadIdx.x < 2 * NUM_DOMAINS) ws[threadIdx.x] = 0;
}

__global__ void count_kernel(const int* __restrict__ y, int* __restrict__ ws, int n) {
    int i = blockIdx.x * blockDim.x + threadIdx.x;
    if (i < n) atomicAdd(&ws[y[i]], 1);
}

__global__ void scatter_kernel(const int* __restrict__ y, int* __restrict__ ws, int n) {
    int i = blockIdx.x * blockDim.x + threadIdx.x;
    if (i >= n) return;
    int dom = y[i];
    int off = 0;
    for (int k = 0; k < NUM_DOMAINS; ++k)       // counts finalized by previous kernel
        if (k < dom) off += ws[k];
    int pos = atomicAdd(&ws[NUM_DOMAINS + dom], 1);
    ws[2 * NUM_DOMAINS + off + pos] = i;
}

// One workgroup (8 waves, 256 threads) computes a 64(M) x 128(N) output tile for
// one domain. Wave (wr, wc) in a 2x4 grid owns a 32x32 sub-tile = 4 WMMA chains.
__global__ __launch_bounds__(256)
void style_gemm_kernel(const float* __restrict__ h,
                       const float* __restrict__ W,
                       const float* __restrict__ bias,
                       const int*   __restrict__ ws,
                       float*       __restrict__ out) {
    __shared__ float As[BM * A_STRIDE];
    __shared__ float Bs[STYLE_DIM * B_STRIDE];
    __shared__ int   rowIdx[BM];
    __shared__ int   rowValid[BM];

    // Map blockIdx.x -> (domain, tile-within-domain) from the device-side counts.
    int tile = blockIdx.x;
    int dom = -1, tloc = 0, off = 0, cnt = 0;
    {
        int trun = 0, pre = 0;
        for (int k = 0; k < NUM_DOMAINS; ++k) {
            int ck = ws[k];
            int tk = (ck + BM - 1) >> 6;
            if (dom < 0 && tile < trun + tk) { dom = k; tloc = tile - trun; off = pre; cnt = ck; }
            trun += tk; pre += ck;
        }
    }
    if (dom < 0) return;   // block-uniform exit (tail blocks)

    const int* idx = ws + 2 * NUM_DOMAINS;
    const int  t   = threadIdx.x;

    if (t < BM) {
        int pos = tloc * BM + t;
        int p2  = pos < cnt ? pos : (cnt - 1);   // clamp loads for partial tiles
        rowIdx[t]   = idx[off + p2];
        rowValid[t] = (pos < cnt) ? 1 : 0;
    }
    __syncthreads();

    const int wave  = t >> 5;
    const int lane  = t & 31;
    const int wr    = wave >> 2;           // 0..1 : row half   (32 rows)
    const int wc    = wave & 3;            // 0..3 : col quarter (32 cols)
    const int lrow  = lane & 15;           // M (for A) / N (for B,C) within a 16-tile
    const int khalf = (lane >> 4) << 1;    // K offset: lanes 0-15 -> {0,1}, 16-31 -> {2,3}

    v8f c00 = {}, c01 = {}, c10 = {}, c11 = {};

    const float* Wd = W + (size_t)dom * STYLE_DIM * AE_DIM;

    for (int k0 = 0; k0 < AE_DIM; k0 += BK) {
        __syncthreads();
        // Stage A: 64 gathered rows x BK floats (1024 float4s; 4 per thread)
#pragma unroll
        for (int it = 0; it < 4; ++it) {
            int q = it * 256 + t;
            int r = q >> 4, c4 = q & 15;
            float4 av = *(const float4*)(h + (size_t)rowIdx[r] * AE_DIM + k0 + c4 * 4);
            *(float4*)(&As[r * A_STRIDE + c4 * 4]) = av;
        }
        // Stage B: 128 contiguous W rows x BK floats (2048 float4s; 8 per thread)
#pragma unroll
        for (int it = 0; it < 8; ++it) {
            int q   = it * 256 + t;
            int col = q >> 4, c4 = q & 15;
            float4 wv = *(const float4*)(Wd + (size_t)col * AE_DIM + k0 + c4 * 4);
            *(float4*)(&Bs[col * B_STRIDE + c4 * 4]) = wv;
        }
        // Prefetch next K-slab of the gathered h rows (W stays L2-resident).
        if (k0 + BK < AE_DIM) {
            int pr = t >> 2, pc = (t & 3) << 4;   // 4 threads/row cover 256B
            __builtin_prefetch(h + (size_t)rowIdx[pr] * AE_DIM + k0 + BK + pc, 0, 3);
        }
        __syncthreads();

#pragma unroll
        for (int ks = 0; ks < BK / 4; ++ks) {
            const int ko = ks * 4 + khalf;
            float2 af0 = *(const float2*)(&As[(wr * 32      + lrow) * A_STRIDE + ko]);
            float2 af1 = *(const float2*)(&As[(wr * 32 + 16 + lrow) * A_STRIDE + ko]);
            float2 bf0 = *(const float2*)(&Bs[(wc * 32      + lrow) * B_STRIDE + ko]);
            float2 bf1 = *(const float2*)(&Bs[(wc * 32 + 16 + lrow) * B_STRIDE + ko]);
            v2f a0 = { af0.x, af0.y }, a1 = { af1.x, af1.y };
            v2f b0 = { bf0.x, bf0.y }, b1 = { bf1.x, bf1.y };
            c00 = __builtin_amdgcn_wmma_f32_16x16x4_f32(false, a0, false, b0, (short)0, c00, false, false);
            c01 = __builtin_amdgcn_wmma_f32_16x16x4_f32(false, a0, false, b1, (short)0, c01, false, false);
            c10 = __builtin_amdgcn_wmma_f32_16x16x4_f32(false, a1, false, b0, (short)0, c10, false, false);
            c11 = __builtin_amdgcn_wmma_f32_16x16x4_f32(false, a1, false, b1, (short)0, c11, false, false);
        }
    }

    // Epilogue: bias add + scatter-store to original row positions.
#pragma unroll
    for (int i = 0; i < 2; ++i) {
#pragma unroll
        for (int j = 0; j < 2; ++j) {
            int   nloc = wc * 32 + j * 16 + lrow;
            float bv   = bias[dom * STYLE_DIM + nloc];
            v8f   cc   = i ? (j ? c11 : c10) : (j ? c01 : c00);
#pragma unroll
            for (int ii = 0; ii < 8; ++ii) {
                // C/D layout: VGPR ii -> M=ii (lanes 0-15), M=ii+8 (lanes 16-31)
                int m = wr * 32 + i * 16 + ((lane >> 4) << 3) + ii;
                if (rowValid[m])
                    out[(size_t)rowIdx[m] * STYLE_DIM + nloc] = cc[ii] + bv;
            }
        }
    }
}

extern "C" void kernel_launch(void* const* d_in, const int* in_sizes, int n_in,
                              void* d_out, int out_size, void* d_ws, size_t ws_size,
                              hipStream_t stream) {
    const float* h    = (const float*)d_in[0];   // [B, 2048] f32
    const int*   y    = (const int*)  d_in[1];   // [B] domain ids
    const float* W    = (const float*)d_in[2];   // [8, 128, 2048] f32
    const float* bias = (const float*)d_in[3];   // [8, 128] f32
    float*       out  = (float*)d_out;           // [B, 128] f32
    int*         ws   = (int*)d_ws;              // needs (16 + B) ints

    zero_counters_kernel<<<1, 32, 0, stream>>>(ws);
    count_kernel  <<<(B_SAMPLES + 255) / 256, 256, 0, stream>>>(y, ws, B_SAMPLES);
    scatter_kernel<<<(B_SAMPLES + 255) / 256, 256, 0, stream>>>(y, ws, B_SAMPLES);

    // 256 full 64-row tiles + up to 8 partial tiles (one per domain)
    style_gemm_kernel<<<B_SAMPLES / BM + NUM_DOMAINS, 256, 0, stream>>>(h, W, bias, ws, out);
}